// RGCN_57380763074878
// MI455X (gfx1250) — compile-verified
//
#include <hip/hip_runtime.h>

// Problem constants (from reference)
#define NN   50000
#define EEDG 800000
#define FIN  128
#define DD   64
#define RR   8
#define LLAY 3

typedef __attribute__((ext_vector_type(2))) float v2f;
typedef __attribute__((ext_vector_type(8))) float v8f;

#define LDSB_STRIDE 72   // floats per B row in LDS (64 + 8 pad: keeps the two
                         // wave halves on disjoint bank halves, 16B-aligned rows)

// ---------------------------------------------------------------------------
// GEMM inner part with async-staged B (GLOBAL_LOAD_ASYNC_TO_LDS_B128).
// LDS layout: ldsF[k][n] row-major, row stride LDSB_STRIDE floats.
// K must be a multiple of 64 (here: 128, 512, 64).
// ---------------------------------------------------------------------------
__device__ __forceinline__ void gemm_part_lds(const float* __restrict__ A, int lda,
                                              const float* __restrict__ B, int K,
                                              float* __restrict__ ldsF,
                                              int m, int half, int ml, v8f* c)
{
    const int tid = threadIdx.x;
    // LDS byte offset of the staging buffer: flat->LDS mapping is addr[31:0].
    const unsigned ldsbase = (unsigned)(uintptr_t)ldsF;

    for (int kb = 0; kb < K; kb += 64) {
        __syncthreads();                        // previous chunk fully consumed
        // Async stage: 64 (K) x 64 (N) floats straight into LDS, no VGPR hop.
        // 256 threads x 4 iters x b128 = 16 KB chunk; ASYNCcnt-tracked.
        #pragma unroll
        for (int it = 0; it < 4; ++it) {
            const int kl = (tid >> 4) + 16 * it;        // 0..63
            const int n4 = (tid & 15) * 4;              // 0,4,...,60
            const float* gp = B + (size_t)(kb + kl) * 64 + n4;
            const unsigned lp = ldsbase + (unsigned)(kl * LDSB_STRIDE + n4) * 4u;
            asm volatile("global_load_async_to_lds_b128 %0, %1, off"
                         :: "v"(lp), "v"(gp) : "memory");
        }
        asm volatile("s_wait_asynccnt 0x0" ::: "memory");
        __syncthreads();

        const float* Arow = A + (size_t)m * lda + kb;
        #pragma unroll 4
        for (int s = 0; s < 16; ++s) {          // 16 K-steps of 4
            const int kk = 4 * s + 2 * half;    // ISA f32 A layout: half picks K pair
            v2f a;
            a.x = Arow[kk];
            a.y = Arow[kk + 1];
            #pragma unroll
            for (int t = 0; t < 4; ++t) {       // 4 column tiles of 16
                const int col = 16 * t + ml;
                v2f b;
                b.x = ldsF[(kk)     * LDSB_STRIDE + col];
                b.y = ldsF[(kk + 1) * LDSB_STRIDE + col];
                c[t] = __builtin_amdgcn_wmma_f32_16x16x4_f32(
                    false, a, false, b, (short)0, c[t], false, false);
            }
        }
    }
}

// ---------------------------------------------------------------------------
// Fused GEMM + bias + ReLU:
//   out[m, 0:64] = relu( A1[m,0:K1] @ B1 + (A2 ? A2[m,0:K2] @ B2 : 0) + bias )
// One wave -> 16x64 strip via V_WMMA_F32_16X16X4_F32.
// No early returns: out-of-range waves clamp A rows and skip only the store,
// so all block barriers are reached and EXEC is all-ones at every WMMA and
// every async LDS stage.
// ---------------------------------------------------------------------------
__global__ __launch_bounds__(256)
void k_gemm_bias_relu(const float* __restrict__ A1, int lda1,
                      const float* __restrict__ B1, int K1,
                      const float* __restrict__ A2, int lda2,
                      const float* __restrict__ B2, int K2,
                      const float* __restrict__ bias,
                      float* __restrict__ out, int M)
{
    __shared__ float ldsF[64 * LDSB_STRIDE];    // 18 KB staged B chunk
    const int wave = threadIdx.x >> 5;
    const int lane = threadIdx.x & 31;
    const int half = lane >> 4;                 // 0: lanes 0-15, 1: lanes 16-31
    const int ml   = lane & 15;
    const int m0   = blockIdx.x * 128 + wave * 16;
    int m = m0 + ml;                            // A row for this lane
    if (m >= M) m = M - 1;                      // clamp (results discarded)

    v8f c[4] = {{}, {}, {}, {}};

    gemm_part_lds(A1, lda1, B1, K1, ldsF, m, half, ml, c);
    if (A2)                                     // kernarg-uniform branch
        gemm_part_lds(A2, lda2, B2, K2, ldsF, m, half, ml, c);

    if (m0 < M) {                               // wave-uniform store guard
        // C/D layout: VGPR j -> row m0 + j + 8*half, col 16*t + ml
        #pragma unroll
        for (int t = 0; t < 4; ++t) {
            const int col = 16 * t + ml;
            const float bv = bias[col];
            #pragma unroll
            for (int j = 0; j < 8; ++j) {
                const float v = c[t][j] + bv;
                out[(size_t)(m0 + j + 8 * half) * 64 + col] = fmaxf(v, 0.0f);
            }
        }
    }
}

// ---------------------------------------------------------------------------
// Zero fill (float4 vectorized)
// ---------------------------------------------------------------------------
__global__ void k_zero(float* __restrict__ p, size_t n)
{
    size_t i = ((size_t)blockIdx.x * blockDim.x + threadIdx.x) * 4;
    if (i + 3 < n) {
        *(float4*)(p + i) = make_float4(0.f, 0.f, 0.f, 0.f);
    } else {
        for (size_t j = i; j < n; ++j) p[j] = 0.f;
    }
}

// ---------------------------------------------------------------------------
// Edge scatter: accum[(dst*R + etype)*64 + d] += h[src*64 + d]; cnt[seg] += 1
// 64 threads per edge (one per feature), 4 edges per 256-thread block.
// This is the bandwidth-bound stage: ~0.5 GB of irregular traffic per layer.
// ---------------------------------------------------------------------------
__global__ __launch_bounds__(256)
void k_scatter(const float* __restrict__ h,
               const int* __restrict__ src, const int* __restrict__ dst,
               const float* __restrict__ eattr,
               float* __restrict__ accum, float* __restrict__ cnt, int E)
{
    const int e = blockIdx.x * 4 + (threadIdx.x >> 6);
    const int d = threadIdx.x & 63;
    if (e >= E) return;
    // Stream-ahead hints for the edge arrays (lowers to global_prefetch).
    if (d == 0 && e + 2048 < E) {
        __builtin_prefetch(src + e + 2048, 0, 0);
        __builtin_prefetch(dst + e + 2048, 0, 0);
        __builtin_prefetch(eattr + (size_t)(e + 2048) * 2, 0, 0);
    }
    const int s = src[e];
    const int t = dst[e];
    const int r = (int)eattr[(size_t)e * 2 + 1];
    const size_t seg = (size_t)t * RR + r;
    atomicAdd(accum + seg * DD + d, h[(size_t)s * DD + d]);
    if (d == 0) atomicAdd(cnt + seg, 1.0f);
}

// ---------------------------------------------------------------------------
// Normalize: accum[i] /= max(cnt[i/64], 1)
// ---------------------------------------------------------------------------
__global__ void k_mean(float* __restrict__ accum, const float* __restrict__ cnt, size_t total)
{
    size_t i = (size_t)blockIdx.x * blockDim.x + threadIdx.x;
    if (i >= total) return;
    float c = cnt[i >> 6];
    accum[i] = accum[i] / fmaxf(c, 1.0f);
}

// ---------------------------------------------------------------------------
// Copy edge distances: out[e] = edge_attr[e*2 + 0]
// ---------------------------------------------------------------------------
__global__ void k_edgedist(const float* __restrict__ eattr, float* __restrict__ out, int E)
{
    int e = blockIdx.x * blockDim.x + threadIdx.x;
    if (e < E) out[e] = eattr[(size_t)e * 2];
}

// ---------------------------------------------------------------------------
extern "C" void kernel_launch(void* const* d_in, const int* in_sizes, int n_in,
                              void* d_out, int out_size, void* d_ws, size_t ws_size,
                              hipStream_t stream)
{
    const float* x      = (const float*)d_in[0];   // (N, 128)
    const int*   ei     = (const int*)  d_in[1];   // (2, E)
    const float* eattr  = (const float*)d_in[2];   // (E, 2)
    const float* W_in   = (const float*)d_in[3];   // (128, 64)
    const float* b_in   = (const float*)d_in[4];   // (64,)
    const float* W_rel  = (const float*)d_in[5];   // (L, R, 64, 64) -> (L, 512, 64)
    const float* W_root = (const float*)d_in[6];   // (L, 64, 64)
    const float* b_conv = (const float*)d_in[7];   // (L, 64)
    float* out = (float*)d_out;                    // h (N*64) ++ edge_distance (E)

    // Workspace layout (floats): h_cur | h_nxt | accum (N*R*64) | cnt (N*R)
    float* h_cur = (float*)d_ws;
    float* h_nxt = h_cur + (size_t)NN * DD;
    float* accum = h_nxt + (size_t)NN * DD;
    float* cnt   = accum + (size_t)NN * RR * DD;

    const dim3 blk(256);
    const int gemm_grid = (NN + 127) / 128;
    const size_t zero_total = (size_t)NN * RR * DD + (size_t)NN * RR; // accum ++ cnt
    const size_t mean_total = (size_t)NN * RR * DD;

    // h = relu(x @ W_in + b_in)
    k_gemm_bias_relu<<<gemm_grid, blk, 0, stream>>>(
        x, FIN, W_in, FIN, nullptr, 0, nullptr, 0, b_in, h_cur, NN);

    const int* src = ei;
    const int* dst = ei + EEDG;

    for (int l = 0; l < LLAY; ++l) {
        k_zero<<<(unsigned)((zero_total / 4 + 255) / 256), blk, 0, stream>>>(accum, zero_total);
        k_scatter<<<(EEDG + 3) / 4, blk, 0, stream>>>(h_cur, src, dst, eattr, accum, cnt, EEDG);
        k_mean<<<(unsigned)((mean_total + 255) / 256), blk, 0, stream>>>(accum, cnt, mean_total);

        float* o = (l == LLAY - 1) ? out : h_nxt;
        // out = relu( mean(N,512) @ Wrel_l(512,64) + h(N,64) @ Wroot_l(64,64) + b_l )
        k_gemm_bias_relu<<<gemm_grid, blk, 0, stream>>>(
            accum, RR * DD, W_rel + (size_t)l * RR * DD * DD, RR * DD,
            h_cur, DD,      W_root + (size_t)l * DD * DD,     DD,
            b_conv + (size_t)l * DD, o, NN);

        float* tmp = h_cur; h_cur = h_nxt; h_nxt = tmp;
    }

    k_edgedist<<<(EEDG + 255) / 256, blk, 0, stream>>>(eattr, out + (size_t)NN * DD, EEDG);
}